// PolynomialRegressor_27152783246050
// MI455X (gfx1250) — compile-verified
//
#include <hip/hip_runtime.h>
#include <hip/hip_bf16.h>

// ---------------------------------------------------------------------------
// PolynomialRegressor on MI455X (gfx1250):
//   out[b,o] = sum_p feat(x[b], p) * W[p,o] + bias[o]
// 47904 monomials of degree 1..3 over 64 inputs, GEMM'd against W[47904,10].
//
// Exact-fp32 WMMA (V_WMMA_F32_16X16X4_F32). Every monomial is factored as
//   feat = P2[a] * X1[b]
// where P2 = per-row table {1.0, all 2080 pairwise products x_i*x_j} staged
// in LDS once per 16-row batch tile, and X1 = {x_0..x_63, 1.0}: A-fragment
// generation costs 2 ds_loads + 1 mul per element.
//
// 256 threads/block = 8 waves = 2 waves per SIMD32 so the DS-latency-bound
// inner loop always has a co-resident wave to issue from.
// ---------------------------------------------------------------------------

typedef __attribute__((ext_vector_type(2))) float v2f;
typedef __attribute__((ext_vector_type(8))) float v8f;

#define N_IN     64
#define N_OUT    10
#define BATCH    2048
#define NPRED    47904              // 64 + 2080 + 45760
#define NPAIR    2080               // C(65,2)
#define NWAVE    8
#define NTHREAD  (NWAVE * 32)
#define KCHUNK   (NPRED / NWAVE)    // 5988 contiguous features per wave
#define KITERS   (KCHUNK / 4)       // 1497 WMMAs per wave (= 3 * 499)
#define XSTRIDE  65                 // 64 inputs + ones column  (==1 mod 64)
#define P2STRIDE 2081               // 1.0 slot + 2080 pairs    (==33 mod 64)

// --- Setup: feature table a|b<<12 and pair table i|j<<8 ---------------------
__global__ __launch_bounds__(256) void poly_build_idx(
    unsigned int* __restrict__ idx,    // [NPRED]  a | (b << 12)
    unsigned int* __restrict__ pairs)  // [NPAIR]  i | (j << 8)
{
    int p = blockIdx.x * 256 + threadIdx.x;

    if (p < NPAIR) {                    // unrank pair (i,j), i<=j
        int q = p, i = 0;
        while (q >= 64 - i) { q -= 64 - i; ++i; }
        pairs[p] = (unsigned)i | ((unsigned)(i + q) << 8);
    }
    if (p >= NPRED) return;

    unsigned a, b;
    if (p < 64) {                       // degree 1: 1.0 * x_p
        a = 0; b = (unsigned)p;
    } else if (p < 64 + NPAIR) {        // degree 2: p2 * 1.0
        a = 1u + (unsigned)(p - 64); b = 64u;
    } else {                            // degree 3: p2(i,j) * x_k
        int q = p - (64 + NPAIR);
        int i = 0;
        for (;;) {
            int cnt = (64 - i) * (65 - i) / 2;
            if (q < cnt) break;
            q -= cnt; ++i;
        }
        int j = i;
        while (q >= 64 - j) { q -= 64 - j; ++j; }
        int k = j + q;
        int rank2 = 64 * i - (i * (i - 1)) / 2 + (j - i);
        a = 1u + (unsigned)rank2; b = (unsigned)k;
    }
    idx[p] = a | (b << 12);
}

// --- Main kernel: 16-row batch tile per block, 8 waves split K --------------
__global__ __launch_bounds__(NTHREAD) void poly_wmma_gemm(
    const float* __restrict__ x,           // [2048, 64]
    const float* __restrict__ W,           // [47904, 10]
    const float* __restrict__ bias,        // [10]
    const unsigned int* __restrict__ idx,  // [47904]
    const unsigned int* __restrict__ pairs,// [2080]
    float* __restrict__ out)               // [2048, 10]
{
    __shared__ float xl[16 * XSTRIDE];     // X1: x tile + ones column (4.1 KB)
    __shared__ float p2[16 * P2STRIDE];    // P2: pairwise products   (133 KB)
    __shared__ float cred[NWAVE * 256];    // per-wave partial C tiles  (8 KB)

    const int t    = threadIdx.x;
    const int lane = t & 31;
    const int wave = __builtin_amdgcn_readfirstlane(t >> 5);
    const int half = (lane >> 4) & 1;      // lanes 16..31
    const int mn   = lane & 15;            // M (A rows) / N (B,C cols)
    const int r0   = blockIdx.x * 16;      // batch-tile origin

    // Stage the 16x64 x-tile + ones column.
    for (int e = t; e < 16 * 64; e += NTHREAD) {
        int row = e >> 6, col = e & 63;
        xl[row * XSTRIDE + col] = x[(r0 + row) * N_IN + col];
    }
    if (t < 16) xl[t * XSTRIDE + 64] = 1.0f;
    __syncthreads();

    // Build P2: all pairwise products for the 16 rows, plus the 1.0 slot.
    for (int pr = t; pr < NPAIR; pr += NTHREAD) {
        unsigned pij = pairs[pr];
        int ci = (int)(pij & 0xffu);
        int cj = (int)((pij >> 8) & 0xffu);
#pragma unroll
        for (int row = 0; row < 16; ++row) {
            p2[row * P2STRIDE + 1 + pr] =
                xl[row * XSTRIDE + ci] * xl[row * XSTRIDE + cj];
        }
    }
    if (t < 16) p2[t * P2STRIDE] = 1.0f;
    __syncthreads();

    // Contiguous K chunk per wave: sequential scalar idx loads + W streaming.
    const int   KOFF   = wave * KCHUNK;
    const int   p2base = mn * P2STRIDE;
    const int   xbase  = mn * XSTRIDE;
    const int   ncol   = (mn < N_OUT) ? mn : (N_OUT - 1);
    const float nmask  = (mn < N_OUT) ? 1.0f : 0.0f;   // zero pad cols 10..15
    const float* __restrict__ wp = W + (KOFF + half * 2) * N_OUT + ncol;
    const unsigned int* __restrict__ ip = idx + KOFF;

    v8f acc0 = {};
    v8f acc1 = {};
    v8f acc2 = {};

    for (int c = 0; c < KITERS; c += 3) {              // 499 iterations
        __builtin_prefetch(wp + (c + 96) * (4 * N_OUT), 0, 1);
#pragma unroll
        for (int u = 0; u < 3; ++u) {
            const int cc = c + u;
            // 4 packed (a,b) indices, wave-uniform address -> scalar load
            // (three consecutive uint4's merge into wide s_load's).
            const uint4 q = *(const uint4*)(ip + cc * 4);
            const unsigned pk0 = half ? q.z : q.x;       // K = 2*half + 0
            const unsigned pk1 = half ? q.w : q.y;       // K = 2*half + 1

            // A fragment: feat = P2[a] * X1[b]  (2 ds_loads + 1 mul each).
            float a0 = p2[p2base + (int)(pk0 & 0xfffu)] * xl[xbase + (int)(pk0 >> 12)];
            float a1 = p2[p2base + (int)(pk1 & 0xfffu)] * xl[xbase + (int)(pk1 >> 12)];

            // B fragment: unconditional coalesced loads, masked pad columns.
            float b0 = wp[cc * (4 * N_OUT)] * nmask;
            float b1 = wp[cc * (4 * N_OUT) + N_OUT] * nmask;

            v2f A = {a0, a1};
            v2f B = {b0, b1};
            if (u == 0)
                acc0 = __builtin_amdgcn_wmma_f32_16x16x4_f32(
                    false, A, false, B, (short)0, acc0, false, false);
            else if (u == 1)
                acc1 = __builtin_amdgcn_wmma_f32_16x16x4_f32(
                    false, A, false, B, (short)0, acc1, false, false);
            else
                acc2 = __builtin_amdgcn_wmma_f32_16x16x4_f32(
                    false, A, false, B, (short)0, acc2, false, false);
        }
    }
    v8f acc = acc0 + acc1 + acc2;

    // Spill per-wave 16x16 partials (C/D layout: VGPR v -> M = v + 8*half).
#pragma unroll
    for (int v = 0; v < 8; ++v) {
        int m = v + half * 8;
        cred[wave * 256 + m * 16 + mn] = acc[v];
    }
    __syncthreads();

    // Cross-wave reduce + bias + store the 10 valid columns.
    {
        int m = t >> 4, n = t & 15;
        if (t < 256 && n < N_OUT) {
            float s = bias[n];
#pragma unroll
            for (int w = 0; w < NWAVE; ++w) s += cred[w * 256 + t];
            out[(r0 + m) * N_OUT + n] = s;
        }
    }
}

extern "C" void kernel_launch(void* const* d_in, const int* in_sizes, int n_in,
                              void* d_out, int out_size, void* d_ws, size_t ws_size,
                              hipStream_t stream) {
    const float* x    = (const float*)d_in[0];   // [2048, 64]
    const float* W    = (const float*)d_in[1];   // [47904, 10]
    const float* bias = (const float*)d_in[2];   // [10]
    float*       out  = (float*)d_out;           // [2048, 10]

    unsigned int* idx   = (unsigned int*)d_ws;        // 47904 u32
    unsigned int* pairs = idx + NPRED;                // 2080 u32 (~196 KB total)

    poly_build_idx<<<(NPRED + 255) / 256, 256, 0, stream>>>(idx, pairs);
    poly_wmma_gemm<<<BATCH / 16, NTHREAD, 0, stream>>>(x, W, bias, idx, pairs, out);
}